// FSOhemCELoss_51522427682917
// MI455X (gfx1250) — compile-verified
//
#include <hip/hip_runtime.h>
#include <hip/hip_bf16.h>
#include <stdint.h>

// Problem constants (fixed shapes from the reference)
#define NN      8
#define CC      19
#define HH      512
#define WW      512
#define HWSZ    (HH * WW)          // 262144
#define PP      (NN * HH * WW)     // 2097152 pixels
#define TILE    256
#define NTILES  (PP / TILE)        // 8192
#define HBINS   65536
#define MIN_KEPT 50000u
#define KRANK   (MIN_KEPT - 1u)    // 49999 (0-based rank of kth smallest)
#define IGNORE_LBL 255
#define THRESH_F 0.7f

// ---------------------------------------------------------------------------
// CDNA5 async global->LDS copy (VGLOBAL GLOBAL_LOAD_ASYNC_TO_LDS_B32, GVS mode)
//   mem_addr = SADDR(64) + VADDR(i32) + IOFFSET ; LDS byte addr in vdst VGPR.
// Tracked by ASYNCcnt; we drain with s_wait_asynccnt 0.
// ---------------------------------------------------------------------------
__device__ __forceinline__ void async_f32_to_lds(unsigned lds_byte_off,
                                                 unsigned gbl_byte_off,
                                                 const float* base) {
  asm volatile("global_load_async_to_lds_b32 %0, %1, %2"
               :
               : "v"(lds_byte_off), "v"(gbl_byte_off), "s"(base)
               : "memory");
}

__device__ __forceinline__ void wait_async0() {
#if __has_builtin(__builtin_amdgcn_s_wait_asynccnt)
  __builtin_amdgcn_s_wait_asynccnt(0);
#else
  asm volatile("s_wait_asynccnt 0" ::: "memory");
#endif
}

// ---------------------------------------------------------------------------
// Kernel 0: zero histograms + accumulators (hist1 | hist2 | misc contiguous)
// ---------------------------------------------------------------------------
__global__ void init_kernel(unsigned* __restrict__ h) {
  unsigned i = blockIdx.x * blockDim.x + threadIdx.x;
  if (i < HBINS * 2u + 16u) h[i] = 0u;
}

// ---------------------------------------------------------------------------
// Kernel 1: per-pixel log-softmax via async-LDS staged class rows.
// Block = 256 pixels; LDS tile = 19 rows x 256 floats (19456 B, offset 0).
// Each lane issues 19 async b32 loads (one per class) into its own LDS
// column and later reads only that column -> ASYNCcnt wait suffices,
// no workgroup barrier required.
// ---------------------------------------------------------------------------
__global__ __launch_bounds__(TILE) void softmax_pass(
    const float* __restrict__ predict, const int* __restrict__ target,
    float* __restrict__ prob, float* __restrict__ loss) {
  __shared__ float tile_lds[CC * TILE];  // sole LDS object -> static offset 0
  const unsigned tid = threadIdx.x;
  const unsigned p   = blockIdx.x * TILE + tid;      // pixel id, grid exactly covers PP
  const unsigned n   = p / HWSZ;
  const unsigned hw  = p - n * HWSZ;
  const unsigned gbase = (n * (unsigned)CC * HWSZ + hw) * 4u;  // < 2^31, fits u32

#pragma unroll
  for (int c = 0; c < CC; ++c) {
    async_f32_to_lds((unsigned)(c * TILE + tid) * 4u,
                     gbase + (unsigned)c * (HWSZ * 4u), predict);
  }
  wait_async0();

  volatile const float* vt = tile_lds;  // force real ds_load of asm-produced data

  float m = -3.402823466e38f;
#pragma unroll
  for (int c = 0; c < CC; ++c) m = fmaxf(m, vt[c * TILE + tid]);

  float s = 0.0f;
#pragma unroll
  for (int c = 0; c < CC; ++c) s += __expf(vt[c * TILE + tid] - m);

  const int t    = target[p];
  const bool ok  = (t != IGNORE_LBL);
  const int  st  = ok ? t : 0;
  const float xt = vt[st * TILE + tid];          // dynamic LDS index (per-lane class)

  const float logp = (xt - m) - __logf(s);
  const float pr   = __expf(logp);

  prob[p] = ok ? pr : 4.0f;   // 4.0 sentinel: sorts above every real prob (<=1.0)
  loss[p] = -logp;
}

// ---------------------------------------------------------------------------
// Kernel 2: histogram of top 16 bits of prob (IEEE bits; prob >= 0 so uint
// order == float order). float4 streaming loads.
// ---------------------------------------------------------------------------
__global__ void hist_top_kernel(const float* __restrict__ prob,
                                unsigned* __restrict__ hist1) {
  const unsigned stride = gridDim.x * blockDim.x;
  const float4* p4 = reinterpret_cast<const float4*>(prob);
  for (unsigned j = blockIdx.x * blockDim.x + threadIdx.x; j < PP / 4; j += stride) {
    float4 v = p4[j];
    atomicAdd(&hist1[__float_as_uint(v.x) >> 16], 1u);
    atomicAdd(&hist1[__float_as_uint(v.y) >> 16], 1u);
    atomicAdd(&hist1[__float_as_uint(v.z) >> 16], 1u);
    atomicAdd(&hist1[__float_as_uint(v.w) >> 16], 1u);
  }
}

// ---------------------------------------------------------------------------
// Kernel 3: find bucket containing rank KRANK + residual rank; also num_valid.
// misc: [0]=num_valid [1]=bucket [2]=residual rank [4]=threshold(f32)
//       [5]=kept count [6]=loss sum(f32)
// ---------------------------------------------------------------------------
__global__ __launch_bounds__(256) void select_top_kernel(
    const unsigned* __restrict__ hist1, unsigned* __restrict__ misc) {
  __shared__ unsigned pre[256];
  const unsigned tid = threadIdx.x;
  unsigned local = 0;
  for (int i = 0; i < 256; ++i) local += hist1[tid * 256 + i];
  pre[tid] = local;
  __syncthreads();
  if (tid == 0) {
    unsigned run = 0;
    for (int i = 0; i < 256; ++i) { unsigned v = pre[i]; pre[i] = run; run += v; }
  }
  __syncthreads();
  const unsigned before = pre[tid];
  if (before <= KRANK && KRANK < before + local) {
    unsigned run = before;
    for (int i = 0; i < 256; ++i) {
      unsigned hv = hist1[tid * 256 + i];
      if (KRANK < run + hv) { misc[1] = tid * 256 + i; misc[2] = KRANK - run; break; }
      run += hv;
    }
  }
  if (tid == 0) misc[0] = PP - hist1[0x4080];  // 4.0f sentinel lands in bin 0x4080
}

// ---------------------------------------------------------------------------
// Kernel 4: low-16-bit histogram restricted to the selected bucket.
// ---------------------------------------------------------------------------
__global__ void hist_low_kernel(const float* __restrict__ prob,
                                const unsigned* __restrict__ misc,
                                unsigned* __restrict__ hist2) {
  const unsigned bucket = misc[1];
  const unsigned stride = gridDim.x * blockDim.x;
  const float4* p4 = reinterpret_cast<const float4*>(prob);
  for (unsigned j = blockIdx.x * blockDim.x + threadIdx.x; j < PP / 4; j += stride) {
    float4 v = p4[j];
    unsigned b;
    b = __float_as_uint(v.x); if ((b >> 16) == bucket) atomicAdd(&hist2[b & 0xFFFFu], 1u);
    b = __float_as_uint(v.y); if ((b >> 16) == bucket) atomicAdd(&hist2[b & 0xFFFFu], 1u);
    b = __float_as_uint(v.z); if ((b >> 16) == bucket) atomicAdd(&hist2[b & 0xFFFFu], 1u);
    b = __float_as_uint(v.w); if ((b >> 16) == bucket) atomicAdd(&hist2[b & 0xFFFFu], 1u);
  }
}

// ---------------------------------------------------------------------------
// Kernel 5: exact kth value from low-16 histogram -> threshold.
// ---------------------------------------------------------------------------
__global__ __launch_bounds__(256) void select_low_kernel(
    const unsigned* __restrict__ hist2, unsigned* __restrict__ misc) {
  __shared__ unsigned pre[256];
  const unsigned tid  = threadIdx.x;
  const unsigned rank = misc[2];
  unsigned local = 0;
  for (int i = 0; i < 256; ++i) local += hist2[tid * 256 + i];
  pre[tid] = local;
  __syncthreads();
  if (tid == 0) {
    unsigned run = 0;
    for (int i = 0; i < 256; ++i) { unsigned v = pre[i]; pre[i] = run; run += v; }
  }
  __syncthreads();
  const unsigned before = pre[tid];
  if (before <= rank && rank < before + local) {
    unsigned run = before;
    for (int i = 0; i < 256; ++i) {
      unsigned hv = hist2[tid * 256 + i];
      if (rank < run + hv) {
        const unsigned bits = (misc[1] << 16) | (unsigned)(tid * 256 + i);
        const float kth = __uint_as_float(bits);
        float thr = (kth > THRESH_F) ? kth : THRESH_F;
        if (MIN_KEPT >= misc[0]) thr = 2.0f;  // keep-all-valid branch (excludes 4.0 sentinels)
        ((float*)misc)[4] = thr;
        break;
      }
      run += hv;
    }
  }
}

// ---------------------------------------------------------------------------
// Kernel 6: final OHEM-masked sum + count.
// ---------------------------------------------------------------------------
__global__ __launch_bounds__(256) void reduce_kernel(
    const float* __restrict__ prob, const float* __restrict__ loss,
    unsigned* __restrict__ misc) {
  const float thr = ((const float*)misc)[4];
  const unsigned stride = gridDim.x * blockDim.x;
  const float4* p4 = reinterpret_cast<const float4*>(prob);
  const float4* l4 = reinterpret_cast<const float4*>(loss);
  float ls = 0.0f; unsigned lc = 0;
  for (unsigned j = blockIdx.x * blockDim.x + threadIdx.x; j < PP / 4; j += stride) {
    float4 pv = p4[j]; float4 lv = l4[j];
    if (pv.x <= thr) { ls += lv.x; ++lc; }
    if (pv.y <= thr) { ls += lv.y; ++lc; }
    if (pv.z <= thr) { ls += lv.z; ++lc; }
    if (pv.w <= thr) { ls += lv.w; ++lc; }
  }
  __shared__ float    ssum[256];
  __shared__ unsigned scnt[256];
  const unsigned tid = threadIdx.x;
  ssum[tid] = ls; scnt[tid] = lc;
  __syncthreads();
  for (int s = 128; s > 0; s >>= 1) {
    if (tid < (unsigned)s) { ssum[tid] += ssum[tid + s]; scnt[tid] += scnt[tid + s]; }
    __syncthreads();
  }
  if (tid == 0) {
    atomicAdd(((float*)misc) + 6, ssum[0]);
    atomicAdd(misc + 5, scnt[0]);
  }
}

__global__ void finalize_kernel(const unsigned* __restrict__ misc,
                                float* __restrict__ out) {
  const unsigned cnt = misc[5];
  const float s = ((const float*)misc)[6];
  out[0] = s / (float)(cnt > 0u ? cnt : 1u);
}

// ---------------------------------------------------------------------------
// Host launcher: inputs [0]=predict f32 (8,19,512,512), [1]=target i32 (8,512,512)
// Workspace: prob(8MB) | loss(8MB) | hist1(256KB) | hist2(256KB) | misc(64B)
// ---------------------------------------------------------------------------
extern "C" void kernel_launch(void* const* d_in, const int* in_sizes, int n_in,
                              void* d_out, int out_size, void* d_ws, size_t ws_size,
                              hipStream_t stream) {
  (void)in_sizes; (void)n_in; (void)out_size; (void)ws_size;
  const float* predict = (const float*)d_in[0];
  const int*   target  = (const int*)d_in[1];
  float* out = (float*)d_out;

  char* ws = (char*)d_ws;
  float*    prob  = (float*)ws;
  float*    loss  = (float*)(ws + (size_t)PP * 4u);
  unsigned* hist1 = (unsigned*)(ws + (size_t)PP * 8u);
  unsigned* hist2 = hist1 + HBINS;
  unsigned* misc  = hist2 + HBINS;

  init_kernel<<<(HBINS * 2 + 16 + 255) / 256, 256, 0, stream>>>(hist1);
  softmax_pass<<<NTILES, TILE, 0, stream>>>(predict, target, prob, loss);
  hist_top_kernel<<<1024, 256, 0, stream>>>(prob, hist1);
  select_top_kernel<<<1, 256, 0, stream>>>(hist1, misc);
  hist_low_kernel<<<1024, 256, 0, stream>>>(prob, misc, hist2);
  select_low_kernel<<<1, 256, 0, stream>>>(hist2, misc);
  reduce_kernel<<<1024, 256, 0, stream>>>(prob, loss, misc);
  finalize_kernel<<<1, 1, 0, stream>>>(misc, out);
}